// LVT_16260746182965
// MI455X (gfx1250) — compile-verified
//
#include <hip/hip_runtime.h>

typedef float v2f __attribute__((ext_vector_type(2)));
typedef float v8f __attribute__((ext_vector_type(8)));

#define VOL_D 64
#define NPTS (VOL_D * VOL_D * VOL_D)   // 262144
#define NVIEW 4
#define NCH 32
#define FH 64
#define FW 64

// ---------------------------------------------------------------------------
// Kernel 1: transpose features [B,V,C,H,W] -> [B,V,H,W,C] (channel-last) so
// each bilinear tap reads 32 contiguous channels (8x global_load_b128).
// ---------------------------------------------------------------------------
__global__ __launch_bounds__(256) void LVT_xpose_kernel(
    const float* __restrict__ feat, float* __restrict__ featT) {
  int idx = blockIdx.x * 256 + threadIdx.x;        // over B*V*H*W
  int hw = idx & 4095;                              // h*64 + w
  int bv = idx >> 12;
  const float* src = feat + (size_t)bv * (NCH * FH * FW) + hw;
  float4* dst = (float4*)(featT + (size_t)idx * NCH);
#pragma unroll
  for (int q = 0; q < 8; ++q) {
    float4 t;
    t.x = src[(4 * q + 0) * (FH * FW)];
    t.y = src[(4 * q + 1) * (FH * FW)];
    t.z = src[(4 * q + 2) * (FH * FW)];
    t.w = src[(4 * q + 3) * (FH * FW)];
    dst[q] = t;
  }
}

// ---------------------------------------------------------------------------
// Kernel 2: WMMA projection + bilinear gather + non-temporal output stream.
// One thread = one (b, grid point p); wave = 32 points.
// Projection for 32 points x 4 views done with two v_wmma_f32_16x16x4_f32:
//   D[16 pts x 16 cols] = A[16x4 (x,y,z,1)] x B[4x16], col 3v+j packs
//   [R_v[j][0..2] ; (R_v @ root + T_v)[j]].  12 columns used.
// ---------------------------------------------------------------------------
__device__ __forceinline__ void acc_tap_b128(float acc[NCH], const float* base,
                                             float w) {
  const float4* p4 = (const float4*)base;
#pragma unroll
  for (int q = 0; q < 8; ++q) {
    float4 f = p4[q];
    acc[4 * q + 0] = fmaf(w, f.x, acc[4 * q + 0]);
    acc[4 * q + 1] = fmaf(w, f.y, acc[4 * q + 1]);
    acc[4 * q + 2] = fmaf(w, f.z, acc[4 * q + 2]);
    acc[4 * q + 3] = fmaf(w, f.w, acc[4 * q + 3]);
  }
}

__global__ __launch_bounds__(256) void LVT_unproj_kernel(
    const float* __restrict__ feat, int chanLast,
    const float* __restrict__ Rm, const float* __restrict__ Tm,
    const float* __restrict__ Km, const float* __restrict__ root,
    float* __restrict__ out) {
  __shared__ float xposeLds[8][32][16];   // per-wave 32x16 WMMA result tiles

  const int b = blockIdx.y;
  const int tid = (int)threadIdx.x;
  const int lane = tid & 31;
  const int wv = tid >> 5;
  const int p = blockIdx.x * 256 + tid;

  // world-space point: grid order p = z*4096 + y*64 + x
  const int gx = p & 63, gy = (p >> 6) & 63, gz = p >> 12;
  const float sc = 0.2f / 63.0f;
  const float wx = gx * sc - 0.1f;
  const float wy = gy * sc - 0.1f;
  const float wz = gz * sc - 0.1f;

  // ---- B matrix (4x16 f32): lanes 0-15 hold K=0 (v0), K=1 (v1);
  //      lanes 16-31 hold K=2 (v0), K=3 (v1); N = lane & 15.
  const int n = lane & 15;
  const int vraw = n / 3;
  const int bvalid = (n < 12) ? 1 : 0;
  const int vc = bvalid ? vraw : 0;
  const int jc = n - 3 * vraw;
  const float* rrow = Rm + ((size_t)(b * NVIEW + vc) * 3 + jc) * 3;
  const float r0 = rrow[0], r1 = rrow[1], r2 = rrow[2];
  const float rt0 = root[b * 3 + 0], rt1 = root[b * 3 + 1],
              rt2 = root[b * 3 + 2];
  const float tj = Tm[(b * NVIEW + vc) * 3 + jc];
  const float nT = fmaf(r0, rt0, fmaf(r1, rt1, fmaf(r2, rt2, tj)));
  const float bsel = (float)bvalid;
  v2f bmat;
  bmat.x = ((lane < 16) ? r0 : r2) * bsel;
  bmat.y = ((lane < 16) ? r1 : nT) * bsel;

  // ---- A matrices (16x4 f32): lanes 0-15 rows M=lane hold K=0,1;
  //      lanes 16-31 rows M=lane-16 hold K=2,3.
  const float wz_lo = __shfl(wz, lane & 15, 32);  // wz of partner lane-16
  const float wx_hi = __shfl(wx, lane | 16, 32);  // wx of partner lane+16
  const float wy_hi = __shfl(wy, lane | 16, 32);
  v2f a0, a1;
  a0.x = (lane < 16) ? wx : wz_lo;
  a0.y = (lane < 16) ? wy : 1.0f;
  a1.x = (lane < 16) ? wx_hi : wz;
  a1.y = (lane < 16) ? wy_hi : 1.0f;

  v8f c0 = {};
  v8f c1 = {};
  c0 = __builtin_amdgcn_wmma_f32_16x16x4_f32(false, a0, false, bmat, (short)0,
                                             c0, false, false);
  c1 = __builtin_amdgcn_wmma_f32_16x16x4_f32(false, a1, false, bmat, (short)0,
                                             c1, false, false);

  // ---- transpose D back to per-lane (point) form via LDS.
  // C/D layout: lane n (0-15): M = vgpr, N = n; lane n+16: M = vgpr+8.
  const int rowb = (lane < 16) ? 0 : 8;
#pragma unroll
  for (int r = 0; r < 8; ++r) {
    xposeLds[wv][rowb + r][n] = c0[r];        // points of lanes 0..15
    xposeLds[wv][16 + rowb + r][n] = c1[r];   // points of lanes 16..31
  }
  asm volatile("s_wait_dscnt 0" ::: "memory");

  float Xc[NVIEW], Yc[NVIEW], Zc[NVIEW];
#pragma unroll
  for (int v = 0; v < NVIEW; ++v) {
    Xc[v] = xposeLds[wv][lane][3 * v + 0];
    Yc[v] = xposeLds[wv][lane][3 * v + 1];
    Zc[v] = xposeLds[wv][lane][3 * v + 2];
  }

  // ---- per-view intrinsics (uniform -> scalar loads), with the
  // pixel-normalization ((2u/255+1)*W-1)/2 folded in:
  //   px = X * (fx*64/255) * (1/z) + (cx*64/255 - 0.5)
  float fxs[NVIEW], fys[NVIEW], cxs[NVIEW], cys[NVIEW];
  const float pixsc = 64.0f / 255.0f;
#pragma unroll
  for (int v = 0; v < NVIEW; ++v) {
    const float* kk = Km + (size_t)(b * NVIEW + v) * 9;
    fxs[v] = kk[0] * pixsc;
    fys[v] = kk[4] * pixsc;
    cxs[v] = kk[2] * pixsc - 0.5f;
    cys[v] = kk[5] * pixsc - 0.5f;
  }

  float acc[NCH];
#pragma unroll
  for (int c = 0; c < NCH; ++c) acc[c] = 0.0f;

#pragma unroll
  for (int v = 0; v < NVIEW; ++v) {
    const float zcl = fmaxf(Zc[v], 1e-5f);
    // fast hardware reciprocal (v_rcp_f32, TRANS unit) instead of IEEE divide
    const float inv = __builtin_amdgcn_rcpf(zcl);
    const float px = fmaf(Xc[v] * fxs[v], inv, cxs[v]);
    const float py = fmaf(Yc[v] * fys[v], inv, cys[v]);
    const float x0f = floorf(px), y0f = floorf(py);
    const float ax = px - x0f, ay = py - y0f;
    const float bx = 1.0f - ax, by = 1.0f - ay;
    const bool vx0 = (x0f >= 0.0f) && (x0f <= 63.0f);
    const bool vx1 = (x0f >= -1.0f) && (x0f <= 62.0f);
    const bool vy0 = (y0f >= 0.0f) && (y0f <= 63.0f);
    const bool vy1 = (y0f >= -1.0f) && (y0f <= 62.0f);
    const float w00 = bx * by * ((vx0 && vy0) ? 1.0f : 0.0f);
    const float w10 = ax * by * ((vx1 && vy0) ? 1.0f : 0.0f);
    const float w01 = bx * ay * ((vx0 && vy1) ? 1.0f : 0.0f);
    const float w11 = ax * ay * ((vx1 && vy1) ? 1.0f : 0.0f);
    const int ix0 = (int)fminf(fmaxf(x0f, 0.0f), 63.0f);
    const int iy0 = (int)fminf(fmaxf(y0f, 0.0f), 63.0f);
    const int ix1 = (int)fminf(fmaxf(x0f + 1.0f, 0.0f), 63.0f);
    const int iy1 = (int)fminf(fmaxf(y0f + 1.0f, 0.0f), 63.0f);

    if (chanLast) {
      const float* base = feat + (size_t)((b * NVIEW + v) * (FH * FW)) * NCH;
      acc_tap_b128(acc, base + (size_t)(iy0 * FW + ix0) * NCH, w00);
      acc_tap_b128(acc, base + (size_t)(iy0 * FW + ix1) * NCH, w10);
      acc_tap_b128(acc, base + (size_t)(iy1 * FW + ix0) * NCH, w01);
      acc_tap_b128(acc, base + (size_t)(iy1 * FW + ix1) * NCH, w11);
    } else {
      const float* base = feat + (size_t)(b * NVIEW + v) * NCH * (FH * FW);
#pragma unroll
      for (int c = 0; c < NCH; ++c) {
        const float* bc = base + (size_t)c * (FH * FW);
        float s = w00 * bc[iy0 * FW + ix0];
        s = fmaf(w10, bc[iy0 * FW + ix1], s);
        s = fmaf(w01, bc[iy1 * FW + ix0], s);
        s = fmaf(w11, bc[iy1 * FW + ix1], s);
        acc[c] += s;
      }
    }
  }

  // ---- mean over views, stream out non-temporally (write-once 64 MiB)
  float* op = out + (size_t)b * NCH * NPTS + p;
#pragma unroll
  for (int c = 0; c < NCH; ++c) {
    __builtin_nontemporal_store(acc[c] * 0.25f, op + (size_t)c * NPTS);
  }
}

extern "C" void kernel_launch(void* const* d_in, const int* in_sizes, int n_in,
                              void* d_out, int out_size, void* d_ws,
                              size_t ws_size, hipStream_t stream) {
  (void)n_in;
  (void)out_size;
  const float* feat = (const float*)d_in[0];
  const float* Rm = (const float*)d_in[1];
  const float* Tm = (const float*)d_in[2];
  const float* Km = (const float*)d_in[3];
  const float* root = (const float*)d_in[4];
  float* out = (float*)d_out;

  const int B = in_sizes[4] / 3;  // root_world is [B,3]
  const size_t needT = (size_t)B * NVIEW * FH * FW * NCH * sizeof(float);
  const int chanLast = (ws_size >= needT) ? 1 : 0;
  float* featT = (float*)d_ws;

  if (chanLast) {
    const int nPix = B * NVIEW * FH * FW;
    LVT_xpose_kernel<<<dim3(nPix / 256), dim3(256), 0, stream>>>(feat, featT);
  }
  LVT_unproj_kernel<<<dim3(NPTS / 256, B), dim3(256), 0, stream>>>(
      chanLast ? featT : feat, chanLast, Rm, Tm, Km, root, out);
}